// MemoryWriteOperation_31499290149008
// MI455X (gfx1250) — compile-verified
//
#include <hip/hip_runtime.h>
#include <hip/hip_bf16.h>
#include <math.h>

// ---------------------------------------------------------------------------
// MemoryWriteOperation for MI455X (gfx1250, wave32).
// Memory-bound (~300MB vs ~75 GFLOP) -> f32 end-to-end, V_WMMA_F32_16X16X4_F32
// for all GEMMs, async global->LDS (ASYNCcnt) double-buffered staging so the
// HBM stream overlaps the matrix pipe.
// ---------------------------------------------------------------------------

typedef __attribute__((ext_vector_type(2))) float v2f;
typedef __attribute__((ext_vector_type(8))) float v8f;

#define D_DIM 512
#define B_DIM 256
#define M_DIM 96
#define W_DIM 128

// Low 32 bits of a generic pointer to __shared__ = LDS byte offset
// (flat aperture).
__device__ __forceinline__ uint32_t lds_addr(const void* p) {
    return (uint32_t)(uintptr_t)p;
}

// Async DMA: memory -> LDS, 16B per lane, tracked by ASYNCcnt.
__device__ __forceinline__ void async_b128(uint32_t lds, const float* g) {
    asm volatile("global_load_async_to_lds_b128 %0, %1, off"
                 :: "v"(lds), "v"(g) : "memory");
}
__device__ __forceinline__ void wait_async_le6() {
    asm volatile("s_wait_asynccnt 0x6" ::: "memory");
}
__device__ __forceinline__ void wait_async_le7() {
    asm volatile("s_wait_asynccnt 0x7" ::: "memory");
}
__device__ __forceinline__ void wait_async_0() {
    asm volatile("s_wait_asynccnt 0x0" ::: "memory");
}

// ===========================================================================
// Kernel 1: Y[N,512] = act( X[N,512] @ W^T + bias ), W row-major [512,512].
// 128x64 output tile per 256-thread workgroup (8 waves, 4x2 wave grid,
// each wave a 32x32 block = 4 WMMA accumulators -> 1 LDS fragment read per
// WMMA). K staged in double-buffered LDS (Kblk=32, pitch 33) via async DMA.
// ===========================================================================
#define G1_BUF (128 * 33 + 64 * 33)   // 6336 floats per buffer

__global__ __launch_bounds__(256)
void gemm_bias_act(const float* __restrict__ X, const float* __restrict__ Wt,
                   const float* __restrict__ bias, float* __restrict__ Y,
                   int N, int act)
{
    // buffer: X tile 128x33 then W tile 64x33 (floats); two buffers.
    __shared__ float smem[2 * G1_BUF];   // 50.7 KB

    const int K    = D_DIM;
    const int row0 = blockIdx.x * 128;
    const int col0 = blockIdx.y * 64;
    const int tid  = threadIdx.x;
    const int lane = tid & 31;
    const int wave = tid >> 5;       // 0..7
    const int rg   = wave & 3;       // rows rg*32
    const int cg   = wave >> 2;      // cols cg*32
    const int hi   = lane >> 4;
    const int lm   = lane & 15;

    // Issue the 6 async b128 loads (per thread) staging one K-block.
    auto issue = [&](int kb, int buf) {
        float* bx = smem + buf * G1_BUF;
        float* bw = bx + 128 * 33;
        #pragma unroll
        for (int i = 0; i < 4; ++i) {                  // 128x32 of X
            int slot = tid + i * 256;                  // 0..1023
            int r = slot >> 3, c4 = (slot & 7) * 4;
            async_b128(lds_addr(bx + r * 33 + c4),
                       X + (size_t)(row0 + r) * K + kb + c4);
        }
        #pragma unroll
        for (int i = 0; i < 2; ++i) {                  // 64x32 of W
            int slot = tid + i * 256;                  // 0..511
            int r = slot >> 3, c4 = (slot & 7) * 4;
            async_b128(lds_addr(bw + r * 33 + c4),
                       Wt + (size_t)(col0 + r) * K + kb + c4);
        }
    };

    v8f acc00 = {}, acc01 = {}, acc10 = {}, acc11 = {};

    issue(0, 0);
    int pb = 0;
    for (int kb = 0; kb < K; kb += 32, pb ^= 1) {
        if (kb + 32 < K) { issue(kb + 32, pb ^ 1); wait_async_le6(); }
        else             { wait_async_0(); }
        __syncthreads();

        const float* xs = smem + pb * G1_BUF;
        const float* ws = xs + 128 * 33;
        #pragma unroll
        for (int kk = 0; kk < 32; kk += 4) {
            // A frags (16x4): lane row = lm; lane-halves carry K pairs.
            v2f a0, a1, b0, b1;
            a0.x = xs[(rg*32 +      lm) * 33 + kk + 2*hi + 0];
            a0.y = xs[(rg*32 +      lm) * 33 + kk + 2*hi + 1];
            a1.x = xs[(rg*32 + 16 + lm) * 33 + kk + 2*hi + 0];
            a1.y = xs[(rg*32 + 16 + lm) * 33 + kk + 2*hi + 1];
            // B frags (4x16): B[k][n] = W[col0+n][kb+kk+k].
            b0.x = ws[(cg*32 +      lm) * 33 + kk + 2*hi + 0];
            b0.y = ws[(cg*32 +      lm) * 33 + kk + 2*hi + 1];
            b1.x = ws[(cg*32 + 16 + lm) * 33 + kk + 2*hi + 0];
            b1.y = ws[(cg*32 + 16 + lm) * 33 + kk + 2*hi + 1];
            acc00 = __builtin_amdgcn_wmma_f32_16x16x4_f32(false, a0, false, b0, (short)0, acc00, false, false);
            acc01 = __builtin_amdgcn_wmma_f32_16x16x4_f32(false, a0, false, b1, (short)0, acc01, false, false);
            acc10 = __builtin_amdgcn_wmma_f32_16x16x4_f32(false, a1, false, b0, (short)0, acc10, false, false);
            acc11 = __builtin_amdgcn_wmma_f32_16x16x4_f32(false, a1, false, b1, (short)0, acc11, false, false);
        }
        __syncthreads();
    }

    // Epilogue: C layout -> VGPR j = row (j + 8*hi), col lm per 16x16 tile.
    const float bi0 = bias[col0 + cg*32 +      lm];
    const float bi1 = bias[col0 + cg*32 + 16 + lm];
    #pragma unroll
    for (int si = 0; si < 2; ++si) {
        v8f cr0 = si ? acc10 : acc00;
        v8f cr1 = si ? acc11 : acc01;
        #pragma unroll
        for (int j = 0; j < 8; ++j) {
            int r = row0 + rg*32 + si*16 + 8*hi + j;
            float v0 = cr0[j] + bi0;
            float v1 = cr1[j] + bi1;
            if (act == 1) {            // sigmoid
                v0 = 1.0f / (1.0f + __expf(-v0));
                v1 = 1.0f / (1.0f + __expf(-v1));
            } else if (act == 2) {     // tanh
                v0 = tanhf(v0);
                v1 = tanhf(v1);
            }
            Y[(size_t)r * K + col0 + cg*32 +      lm] = v0;
            Y[(size_t)r * K + col0 + cg*32 + 16 + lm] = v1;
        }
    }
}

// ===========================================================================
// Kernel 2: per-batch scores = (q @ k^T)/sqrt(D), softmax over M=96, write
// transposed attnT[b][m][w]. One 256-thread WG per batch; wave i owns query
// rows 16i..16i+15 x all 96 keys (6 accumulators). Async double-buffered LDS.
// ===========================================================================
#define G2_BUF (128 * 33 + 96 * 33)   // 7392 floats per buffer

__global__ __launch_bounds__(256)
void attn_softmax(const float* __restrict__ q, const float* __restrict__ kmat,
                  float* __restrict__ attnT)
{
    // phase 1: two buffers of (q 128x33 + k 96x33); phase 2: sc[128][97] alias.
    __shared__ float smem[2 * G2_BUF];   // 59.1 KB >= 128*97

    const int b    = blockIdx.x;
    const int tid  = threadIdx.x;
    const int lane = tid & 31;
    const int wave = tid >> 5;
    const int hi   = lane >> 4;
    const int lm   = lane & 15;

    auto issue = [&](int kb, int buf) {
        float* bq = smem + buf * G2_BUF;
        float* bk = bq + 128 * 33;
        #pragma unroll
        for (int i = 0; i < 4; ++i) {                  // 128x32 of q
            int slot = tid + i * 256;
            int r = slot >> 3, c4 = (slot & 7) * 4;
            async_b128(lds_addr(bq + r * 33 + c4),
                       q + ((size_t)b * W_DIM + r) * D_DIM + kb + c4);
        }
        #pragma unroll
        for (int i = 0; i < 3; ++i) {                  // 96x32 of k
            int slot = tid + i * 256;
            int r = slot >> 3, c4 = (slot & 7) * 4;
            async_b128(lds_addr(bk + r * 33 + c4),
                       kmat + ((size_t)b * M_DIM + r) * D_DIM + kb + c4);
        }
    };

    v8f acc[6] = {};

    issue(0, 0);
    int pb = 0;
    for (int kb = 0; kb < D_DIM; kb += 32, pb ^= 1) {
        if (kb + 32 < D_DIM) { issue(kb + 32, pb ^ 1); wait_async_le7(); }
        else                 { wait_async_0(); }
        __syncthreads();

        const float* qs = smem + pb * G2_BUF;
        const float* ks = qs + 128 * 33;
        #pragma unroll
        for (int kk = 0; kk < 32; kk += 4) {
            v2f a;
            a.x = qs[(wave*16 + lm) * 33 + kk + 2*hi + 0];
            a.y = qs[(wave*16 + lm) * 33 + kk + 2*hi + 1];
            #pragma unroll
            for (int t = 0; t < 6; ++t) {
                v2f bf;
                bf.x = ks[(t*16 + lm) * 33 + kk + 2*hi + 0];
                bf.y = ks[(t*16 + lm) * 33 + kk + 2*hi + 1];
                acc[t] = __builtin_amdgcn_wmma_f32_16x16x4_f32(false, a, false, bf, (short)0, acc[t], false, false);
            }
        }
        __syncthreads();
    }

    // Spill scaled scores to LDS: sc[w][m], pitch 97.
    const float scale = 0.044194173824159216f;  // 1/sqrt(512)
    float* sc = smem;
    #pragma unroll
    for (int t = 0; t < 6; ++t) {
        #pragma unroll
        for (int j = 0; j < 8; ++j) {
            int w = wave*16 + 8*hi + j;
            int m = t*16 + lm;
            sc[w*97 + m] = acc[t][j] * scale;
        }
    }
    __syncthreads();

    // Row softmax over M=96: one thread per query row.
    if (tid < W_DIM) {
        float* row = sc + tid * 97;
        float mx = row[0];
        for (int m = 1; m < M_DIM; ++m) mx = fmaxf(mx, row[m]);
        float s = 0.0f;
        for (int m = 0; m < M_DIM; ++m) { float e = __expf(row[m] - mx); row[m] = e; s += e; }
        float inv = 1.0f / s;
        for (int m = 0; m < M_DIM; ++m) row[m] *= inv;
    }
    __syncthreads();

    // Coalesced transposed store: attnT[b][m][w].
    for (int i = tid; i < M_DIM * W_DIM; i += 256) {
        int m = i >> 7;
        int w = i & 127;
        attnT[((size_t)b * M_DIM + m) * W_DIM + w] = sc[w*97 + m];
    }
}

// ===========================================================================
// Kernel 3: sequential erase/add scan over W tokens + LayerNorm.
// Block = 512 threads (one per d), each block owns 8 memory rows of one batch
// (8x reuse of the e/add streams). attn rows pre-staged in LDS; streaming
// reads prefetched ahead (global_prefetch_b8).
// ===========================================================================
__global__ __launch_bounds__(512)
void update_ln(const float* __restrict__ memory, const float* __restrict__ attnT,
               const float* __restrict__ erase,  const float* __restrict__ addv,
               const float* __restrict__ gamma,  const float* __restrict__ beta,
               float* __restrict__ out)
{
    __shared__ float at[8 * W_DIM];
    __shared__ float red[512];

    const int b  = blockIdx.x;
    const int m0 = blockIdx.y * 8;
    const int d  = threadIdx.x;

    for (int i = d; i < 8 * W_DIM; i += 512) {
        int ml = i >> 7, w = i & 127;
        at[i] = attnT[((size_t)b * M_DIM + m0 + ml) * W_DIM + w];
    }

    float mv[8];
    #pragma unroll
    for (int r = 0; r < 8; ++r)
        mv[r] = memory[((size_t)b * M_DIM + m0 + r) * D_DIM + d];
    __syncthreads();

    const float* eb = erase + (size_t)b * W_DIM * D_DIM;
    const float* ab = addv  + (size_t)b * W_DIM * D_DIM;
    for (int w = 0; w < W_DIM; ++w) {
        if (w + 4 < W_DIM) {
            __builtin_prefetch(eb + (size_t)(w + 4) * D_DIM + d, 0, 0);
            __builtin_prefetch(ab + (size_t)(w + 4) * D_DIM + d, 0, 0);
        }
        float e  = eb[(size_t)w * D_DIM + d];
        float ad = ab[(size_t)w * D_DIM + d];
        #pragma unroll
        for (int r = 0; r < 8; ++r) {
            float a = at[r * W_DIM + w];
            mv[r] = fmaf(-a * e, mv[r], mv[r]) + a * ad;   // mem*(1-a*e)+a*ad
        }
    }

    const float g  = gamma[d];
    const float bt = beta[d];
    #pragma unroll 1
    for (int r = 0; r < 8; ++r) {
        red[d] = mv[r];
        __syncthreads();
        for (int s = 256; s > 0; s >>= 1) {
            if (d < s) red[d] += red[d + s];
            __syncthreads();
        }
        float mu = red[0] * (1.0f / 512.0f);
        __syncthreads();
        float diff = mv[r] - mu;
        red[d] = diff * diff;
        __syncthreads();
        for (int s = 256; s > 0; s >>= 1) {
            if (d < s) red[d] += red[d + s];
            __syncthreads();
        }
        float var = red[0] * (1.0f / 512.0f);
        __syncthreads();
        float inv = rsqrtf(var + 1e-5f);
        out[((size_t)b * M_DIM + m0 + r) * D_DIM + d] = diff * inv * g + bt;
    }
}

// ===========================================================================
// Launch
// ===========================================================================
extern "C" void kernel_launch(void* const* d_in, const int* in_sizes, int n_in,
                              void* d_out, int out_size, void* d_ws, size_t ws_size,
                              hipStream_t stream)
{
    const float* memory = (const float*)d_in[0];
    const float* wtok   = (const float*)d_in[1];
    const float* Wq = (const float*)d_in[2];  const float* bq = (const float*)d_in[3];
    const float* Wk = (const float*)d_in[4];  const float* bk = (const float*)d_in[5];
    const float* We = (const float*)d_in[6];  const float* be = (const float*)d_in[7];
    const float* Wa = (const float*)d_in[8];  const float* ba = (const float*)d_in[9];
    const float* gamma = (const float*)d_in[10];
    const float* beta  = (const float*)d_in[11];
    float* out = (float*)d_out;

    const int TOK_N = B_DIM * W_DIM;   // 32768
    const int MEM_N = B_DIM * M_DIM;   // 24576

    // Workspace layout (floats). Total = 66,060,288 f32 = ~252 MB.
    float* ws    = (float*)d_ws;
    float* q     = ws;
    float* e     = q + (size_t)TOK_N * D_DIM;
    float* a     = e + (size_t)TOK_N * D_DIM;
    float* kproj = a + (size_t)TOK_N * D_DIM;
    float* attnT = kproj + (size_t)MEM_N * D_DIM;

    dim3 blk(256);
    dim3 gTok(TOK_N / 128, D_DIM / 64);
    gemm_bias_act<<<gTok, blk, 0, stream>>>(wtok, Wq, bq, q, TOK_N, 0);
    gemm_bias_act<<<gTok, blk, 0, stream>>>(wtok, We, be, e, TOK_N, 1);
    gemm_bias_act<<<gTok, blk, 0, stream>>>(wtok, Wa, ba, a, TOK_N, 2);
    dim3 gMem(MEM_N / 128, D_DIM / 64);
    gemm_bias_act<<<gMem, blk, 0, stream>>>(memory, Wk, bk, kproj, MEM_N, 0);

    attn_softmax<<<dim3(B_DIM), blk, 0, stream>>>(q, kproj, attnT);

    update_ln<<<dim3(B_DIM, M_DIM / 8), dim3(512), 0, stream>>>(
        memory, attnT, e, a, gamma, beta, out);
}